// GCNemb_42082089566348
// MI455X (gfx1250) — compile-verified
//
#include <hip/hip_runtime.h>

#define N_NODES 50000
#define N_EDGES 400000

typedef float v2f __attribute__((ext_vector_type(2)));
typedef float v8f __attribute__((ext_vector_type(8)));

// ---------------- degree / norm precompute ----------------

__global__ void k_set_deg(float* deg, int n) {
  int i = blockIdx.x * blockDim.x + threadIdx.x;
  if (i < n) deg[i] = 1.0f;  // self loop contributes 1
}

__global__ void k_deg_accum(const int* __restrict__ dst, float* deg, int e) {
  int i = blockIdx.x * blockDim.x + threadIdx.x;
  if (i < e) atomicAdd(&deg[dst[i]], 1.0f);
}

__global__ void k_dinv(float* deg, int n) {
  int i = blockIdx.x * blockDim.x + threadIdx.x;
  if (i < n) deg[i] = rsqrtf(deg[i]);  // deg >= 1 always (self loop)
}

__global__ void k_norm(const int* __restrict__ src, const int* __restrict__ dst,
                       const float* __restrict__ dinv, float* __restrict__ norm, int e) {
  int i = blockIdx.x * blockDim.x + threadIdx.x;
  if (i < e) norm[i] = dinv[src[i]] * dinv[dst[i]];
}

// ---------------- fp32 WMMA GEMM: H[N,FO] = act(X)[N,FI] @ W[FI,FO] ----------------
// Each wave computes a 32x64 strip: 8 accumulators; 2 A fragments (rows m0, m0+16)
// share 4 B fragments per K-step -> 10 VMEM instrs per 8 V_WMMA_F32_16X16X4_F32.
// All lane-varying address components are folded into base pointers computed once;
// every load/store in the K loop uses compile-time immediate offsets.

template <int FI, int FO, bool RELU>
__global__ __launch_bounds__(128) void k_gemm_wmma(const float* __restrict__ X,
                                                   const float* __restrict__ W,
                                                   float* __restrict__ H) {
  const int lane = threadIdx.x & 31;
  const int half = lane >> 4;  // 0 or 1
  const int l16  = lane & 15;
  const int wave = threadIdx.x >> 5;
  const int m0 = blockIdx.y * 128 + wave * 32;
  if (m0 >= N_NODES) return;                 // wave-uniform
  const int  m1   = m0 + 16;
  const bool m1ok = (m1 < N_NODES);          // wave-uniform (50000 % 32 != 0 tail)
  const int  m1c  = m1ok ? m1 : m0;          // clamp: duplicate compute, skip store
  const int  n0   = blockIdx.x * 64;

  // A bases: element kb = k + 2*half of each row
  const float* xr0 = X + (size_t)(m0 + l16) * FI + 2 * half;
  const float* xr1 = X + (size_t)(m1c + l16) * FI + 2 * half;

  // B bases per 16-col tile: loads become wb[j][k*FO] / wb[j][k*FO + FO]
  int col[4];
  const float* wb[4];
#pragma unroll
  for (int j = 0; j < 4; ++j) {
    col[j] = n0 + 16 * j + l16;
    int cc = col[j];
    if constexpr (FO % 64 != 0) cc = (cc < FO) ? cc : (FO - 1);  // only FO==40
    wb[j] = W + (size_t)(2 * half) * FO + cc;
  }

  v8f acc0[4], acc1[4];
#pragma unroll
  for (int j = 0; j < 4; ++j) {
    acc0[j] = v8f{0.f, 0.f, 0.f, 0.f, 0.f, 0.f, 0.f, 0.f};
    acc1[j] = v8f{0.f, 0.f, 0.f, 0.f, 0.f, 0.f, 0.f, 0.f};
  }

  constexpr int KM = FI & ~3;
#pragma unroll 2
  for (int k = 0; k < KM; k += 4) {
    v2f a0, a1;
    {
      float t0 = xr0[k], t1 = xr0[k + 1];
      if (RELU) { t0 = fmaxf(t0, 0.f); t1 = fmaxf(t1, 0.f); }
      a0.x = t0; a0.y = t1;
    }
    {
      float t0 = xr1[k], t1 = xr1[k + 1];
      if (RELU) { t0 = fmaxf(t0, 0.f); t1 = fmaxf(t1, 0.f); }
      a1.x = t0; a1.y = t1;
    }
#pragma unroll
    for (int j = 0; j < 4; ++j) {
      v2f b;
      b.x = wb[j][(size_t)k * FO];
      b.y = wb[j][(size_t)k * FO + FO];
      acc0[j] = __builtin_amdgcn_wmma_f32_16x16x4_f32(false, a0, false, b, (short)0,
                                                      acc0[j], false, false);
      acc1[j] = __builtin_amdgcn_wmma_f32_16x16x4_f32(false, a1, false, b, (short)0,
                                                      acc1[j], false, false);
    }
  }
  if constexpr (KM < FI) {  // K tail: only layer 0 (FI == 3), branchless
    const int  kb = KM + 2 * half;
    const bool v1 = (kb + 1 < FI);  // first element always valid for FI==3
    const int  o1 = v1 ? 1 : 0;     // clamped offset for second element
    v2f a0, a1;
    {
      float t0 = xr0[KM], t1 = xr0[KM + o1];
      if (!v1) t1 = 0.f;
      if (RELU) { t0 = fmaxf(t0, 0.f); t1 = fmaxf(t1, 0.f); }
      a0.x = t0; a0.y = t1;
    }
    {
      float t0 = xr1[KM], t1 = xr1[KM + o1];
      if (!v1) t1 = 0.f;
      if (RELU) { t0 = fmaxf(t0, 0.f); t1 = fmaxf(t1, 0.f); }
      a1.x = t0; a1.y = t1;
    }
#pragma unroll
    for (int j = 0; j < 4; ++j) {
      v2f b;
      b.x = wb[j][(size_t)KM * FO];
      float by = wb[j][(size_t)KM * FO + (v1 ? FO : 0)];
      b.y = v1 ? by : 0.f;
      acc0[j] = __builtin_amdgcn_wmma_f32_16x16x4_f32(false, a0, false, b, (short)0,
                                                      acc0[j], false, false);
      acc1[j] = __builtin_amdgcn_wmma_f32_16x16x4_f32(false, a1, false, b, (short)0,
                                                      acc1[j], false, false);
    }
  }

  // stores: per-tile base pointer + v*FO immediate offsets
  const int rb0 = m0 + half * 8;
  const int rb1 = m1 + half * 8;
#pragma unroll
  for (int j = 0; j < 4; ++j) {
    if (FO % 64 == 0 || col[j] < FO) {
      float* h0 = H + (size_t)rb0 * FO + col[j];
#pragma unroll
      for (int v = 0; v < 8; ++v) h0[(size_t)v * FO] = acc0[j][v];
      if (m1ok) {
        float* h1 = H + (size_t)rb1 * FO + col[j];
#pragma unroll
        for (int v = 0; v < 8; ++v) h1[(size_t)v * FO] = acc1[j][v];
      }
    }
  }
}

// ---------------- aggregation: Y = b + dinv^2 * H  (self loop + bias, full overwrite) ----

template <int FO>
__global__ void k_agg_init(const float* __restrict__ H, const float* __restrict__ bias,
                           const float* __restrict__ dinv, float* __restrict__ Y) {
  constexpr int CPT = FO / 4;
  int tid = blockIdx.x * blockDim.x + threadIdx.x;
  int i  = tid / CPT;
  int c4 = (tid - i * CPT) << 2;
  if (i >= N_NODES) return;
  float dn = dinv[i];
  float sn = dn * dn;
  float4 h = *(const float4*)(H + (size_t)i * FO + c4);
  float4 b = *(const float4*)(bias + c4);
  float4 y;
  y.x = b.x + sn * h.x; y.y = b.y + sn * h.y;
  y.z = b.z + sn * h.z; y.w = b.w + sn * h.w;
  *(float4*)(Y + (size_t)i * FO + c4) = y;
}

// ---------------- aggregation: Y[dst] += norm[e] * H[src]  (float atomics) --------------

template <int FO>
__global__ void k_agg_edges(const float* __restrict__ H, const int* __restrict__ src,
                            const int* __restrict__ dst, const float* __restrict__ norm,
                            float* __restrict__ Y) {
  constexpr int CPT = FO / 4;
  int tid = blockIdx.x * blockDim.x + threadIdx.x;
  int e  = tid / CPT;
  int c4 = (tid - e * CPT) << 2;
  if (e >= N_EDGES) return;
  int s = src[e], d = dst[e];
  float w = norm[e];
  float4 h = *(const float4*)(H + (size_t)s * FO + c4);
  float* y = Y + (size_t)d * FO + c4;
  atomicAdd(y + 0, w * h.x);
  atomicAdd(y + 1, w * h.y);
  atomicAdd(y + 2, w * h.z);
  atomicAdd(y + 3, w * h.w);
}

// ---------------- per-layer driver ----------------

template <int FI, int FO, bool RELU>
static void run_layer(const float* X, const float* W, const float* b, const float* dinv,
                      const float* norm, const int* src, const int* dst, float* H,
                      float* Y, hipStream_t stream) {
  // columns on blockIdx.x (fastest) so all column groups of a row block are
  // dispatched adjacently -> X row tiles stay hot in the 192MB L2.
  dim3 grid((FO + 63) / 64, (N_NODES / 16 + 7) / 8);  // 8 row tiles per block
  k_gemm_wmma<FI, FO, RELU><<<grid, 128, 0, stream>>>(X, W, H);

  constexpr int T = 256;
  constexpr long totn = (long)N_NODES * (FO / 4);
  k_agg_init<FO><<<(int)((totn + T - 1) / T), T, 0, stream>>>(H, b, dinv, Y);
  constexpr long tote = (long)N_EDGES * (FO / 4);
  k_agg_edges<FO><<<(int)((tote + T - 1) / T), T, 0, stream>>>(H, src, dst, norm, Y);
}

// ---------------- host side ----------------

extern "C" void kernel_launch(void* const* d_in, const int* in_sizes, int n_in,
                              void* d_out, int out_size, void* d_ws, size_t ws_size,
                              hipStream_t stream) {
  (void)in_sizes; (void)n_in; (void)out_size; (void)ws_size;

  const float* x  = (const float*)d_in[0];
  const int*   ei = (const int*)d_in[1];
  const float* Wt[8];
  const float* bs[8];
  for (int i = 0; i < 8; ++i) {
    Wt[i] = (const float*)d_in[2 + 2 * i];
    bs[i] = (const float*)d_in[3 + 2 * i];
  }

  // workspace carve (~514 MB): dinv | norm | H(N x 1024) | P0(N x 1024) | P1(N x 512)
  char* ws = (char*)d_ws;
  float* dinv = (float*)ws; ws += ((size_t)N_NODES * 4 + 255) & ~(size_t)255;
  float* norm = (float*)ws; ws += ((size_t)N_EDGES * 4 + 255) & ~(size_t)255;
  float* H    = (float*)ws; ws += (size_t)N_NODES * 1024 * 4;
  float* P0   = (float*)ws; ws += (size_t)N_NODES * 1024 * 4;
  float* P1   = (float*)ws;
  float* out  = (float*)d_out;

  const int* src = ei;
  const int* dst = ei + N_EDGES;
  const int T = 256;

  k_set_deg<<<(N_NODES + T - 1) / T, T, 0, stream>>>(dinv, N_NODES);
  k_deg_accum<<<(N_EDGES + T - 1) / T, T, 0, stream>>>(dst, dinv, N_EDGES);
  k_dinv<<<(N_NODES + T - 1) / T, T, 0, stream>>>(dinv, N_NODES);
  k_norm<<<(N_EDGES + T - 1) / T, T, 0, stream>>>(src, dst, dinv, norm, N_EDGES);

  // DIMS = [3, 64, 64, 64, 128, 1024, 512, 256, 40]
  run_layer<3,    64,   false>(x,  Wt[0], bs[0], dinv, norm, src, dst, H, P0, stream);
  run_layer<64,   64,   true >(P0, Wt[1], bs[1], dinv, norm, src, dst, H, P1, stream);
  run_layer<64,   64,   true >(P1, Wt[2], bs[2], dinv, norm, src, dst, H, P0, stream);
  run_layer<64,   128,  true >(P0, Wt[3], bs[3], dinv, norm, src, dst, H, P1, stream);
  run_layer<128,  1024, true >(P1, Wt[4], bs[4], dinv, norm, src, dst, H, P0, stream);
  run_layer<1024, 512,  true >(P0, Wt[5], bs[5], dinv, norm, src, dst, H, P1, stream);
  run_layer<512,  256,  true >(P1, Wt[6], bs[6], dinv, norm, src, dst, H, P0, stream);
  run_layer<256,  40,   true >(P0, Wt[7], bs[7], dinv, norm, src, dst, H, out, stream);
}